// ScaledDotProductAttention_20753281974874
// MI455X (gfx1250) — compile-verified
//
#include <hip/hip_runtime.h>
#include <hip/hip_bf16.h>

// Scaled dot-product attention, fp32, B=128 S=1024 Dk=64.
// Bandwidth-bound on writing the full attn matrix (537MB) -> keep fp32 and use
// V_WMMA_F32_16X16X4_F32 for both QK^T and attn@V. One workgroup (4 wave32)
// handles one 16-row query tile of one batch.

typedef __attribute__((ext_vector_type(2))) float v2f;
typedef __attribute__((ext_vector_type(4))) float v4f;
typedef __attribute__((ext_vector_type(8))) float v8f;

#define SEQ 1024
#define DK 64
#define BDIM 128
#define NWAVE 4
#define COLS_PER_WAVE (SEQ / NWAVE)   // 256 keys per wave in phase 1
#define LDS_STRIDE 1036               // 1024 + pad (stride%64==12: conflict-free-ish)

__global__ __launch_bounds__(BDIM)
void sdpa_wmma_f32_kernel(const float* __restrict__ qp,
                          const float* __restrict__ kp,
                          const float* __restrict__ vp,
                          const unsigned char* __restrict__ maskp, // jnp.bool_: 1 byte/elem
                          float* __restrict__ outp,   // [B,S,DK]
                          float* __restrict__ attnp)  // [B,S,S]
{
    __shared__ float smem[16 * LDS_STRIDE + 16];
    float* rowscale = &smem[16 * LDS_STRIDE];

    const int tid  = threadIdx.x;
    const int wave = tid >> 5;
    const int lane = tid & 31;
    const int lh   = lane >> 4;   // lane half: selects K pair {0,1} vs {2,3}
    const int ln   = lane & 15;   // row (A) / col (B,C,D) index within tile

    const int b     = blockIdx.x >> 6;        // batch*head
    const int qrow0 = (blockIdx.x & 63) << 4; // 16-row query tile base

    // ---------------- Phase 1: scores = (Q K^T) * 1/sqrt(1024), masked -------
    // A fragments (Q tile 16xDK as 16 steps of 16x4). ISA layout:
    //   lanes 0-15: M=lane, v2f = Q[M][4i+0..1]; lanes 16-31: Q[M][4i+2..3]
    const size_t qbase = ((size_t)b * SEQ + qrow0 + ln) * DK + lh * 2;
    v2f aq[16];
    #pragma unroll
    for (int i = 0; i < 16; ++i)
        aq[i] = *(const v2f*)(qp + qbase + i * 4);

    const int col0 = wave * COLS_PER_WAVE;
    for (int kt = 0; kt < 16; ++kt) {
        const int key0 = col0 + kt * 16;
        // B fragments (K^T sub-block 4x16): lane holds col N=ln,
        //   lanes 0-15: v2f = K[key0+N][4i+0..1]; lanes 16-31: K[key0+N][4i+2..3]
        const size_t kbase = ((size_t)b * SEQ + key0 + ln) * DK + lh * 2;
        v2f bk[16];
        #pragma unroll
        for (int i = 0; i < 16; ++i)
            bk[i] = *(const v2f*)(kp + kbase + i * 4);

        v8f acc = {};
        #pragma unroll
        for (int i = 0; i < 16; ++i)
            acc = __builtin_amdgcn_wmma_f32_16x16x4_f32(
                false, aq[i], false, bk[i], (short)0, acc, false, false);

        // C/D layout: acc[r] = row (r + lh*8), col ln. Scale + mask -> LDS.
        #pragma unroll
        for (int r = 0; r < 8; ++r) {
            const int m = r + lh * 8;
            const unsigned char msk =
                maskp[((size_t)b * SEQ + qrow0 + m) * SEQ + key0 + ln];
            const float val = msk ? -__builtin_inff() : acc[r] * 0.03125f;
            smem[m * LDS_STRIDE + key0 + ln] = val;
        }
    }
    __syncthreads();

    // ---------------- Phase 2: softmax over each of the 16 rows --------------
    {
        const int row = tid >> 3;   // 8 threads per row
        const int j   = tid & 7;
        float* srow = &smem[row * LDS_STRIDE];

        float mx = -__builtin_inff();
        for (int i = 0; i < SEQ / 8; ++i)
            mx = fmaxf(mx, srow[j + 8 * i]);
        mx = fmaxf(mx, __shfl_xor(mx, 1, 32));
        mx = fmaxf(mx, __shfl_xor(mx, 2, 32));
        mx = fmaxf(mx, __shfl_xor(mx, 4, 32));

        float sum = 0.f;
        if (mx == -__builtin_inff()) {
            // fully masked row: reference maps softmax NaNs to zeros
            for (int i = 0; i < SEQ / 8; ++i) srow[j + 8 * i] = 0.f;
        } else {
            for (int i = 0; i < SEQ / 8; ++i) {
                const float e = __expf(srow[j + 8 * i] - mx);
                srow[j + 8 * i] = e;
                sum += e;
            }
        }
        sum += __shfl_xor(sum, 1, 32);
        sum += __shfl_xor(sum, 2, 32);
        sum += __shfl_xor(sum, 4, 32);
        if (j == 0) rowscale[row] = (sum > 0.f) ? (1.f / sum) : 0.f;
    }
    __syncthreads();

    // Normalize in LDS and stream attn tile to global (coalesced float4).
    #pragma unroll
    for (int it = 0; it < 32; ++it) {
        const int f   = tid + it * BDIM;  // float4 index within 16x1024 tile
        const int row = f >> 8;           // 256 float4 per row
        const int c4  = f & 255;
        const float s = rowscale[row];
        v4f val = *(v4f*)(&smem[row * LDS_STRIDE + c4 * 4]);
        val *= s;
        *(v4f*)(&smem[row * LDS_STRIDE + c4 * 4]) = val;
        *(v4f*)(attnp + ((size_t)b * SEQ + qrow0 + row) * SEQ + c4 * 4) = val;
    }
    __syncthreads();

    // ---------------- Phase 3: output = attn @ V ------------------------------
    // Each wave produces output columns [wave*16, wave*16+16) for all 16 rows.
    {
        const int n0 = wave * 16;
        v8f oacc = {};
        const float* vbase = vp + (size_t)b * SEQ * DK + n0 + ln;
        #pragma unroll 4
        for (int kb = 0; kb < SEQ; kb += 4) {
            // A: attn tile rows from LDS (16x4 step)
            const v2f a = *(const v2f*)(&smem[ln * LDS_STRIDE + kb + lh * 2]);
            // B: V sub-block 4x16; lane holds col n0+ln, K rows kb+lh*2, +1
            v2f bb;
            bb.x = vbase[(kb + lh * 2) * DK];
            bb.y = vbase[(kb + lh * 2 + 1) * DK];
            oacc = __builtin_amdgcn_wmma_f32_16x16x4_f32(
                false, a, false, bb, (short)0, oacc, false, false);
        }
        #pragma unroll
        for (int r = 0; r < 8; ++r)
            outp[((size_t)b * SEQ + qrow0 + r + lh * 8) * DK + n0 + ln] = oacc[r];
    }
}

extern "C" void kernel_launch(void* const* d_in, const int* in_sizes, int n_in,
                              void* d_out, int out_size, void* d_ws, size_t ws_size,
                              hipStream_t stream) {
    const float* q = (const float*)d_in[0];
    const float* k = (const float*)d_in[1];
    const float* v = (const float*)d_in[2];
    const unsigned char* mask = (const unsigned char*)d_in[3];

    const int nB = in_sizes[0] / (SEQ * DK);   // 128
    float* out  = (float*)d_out;                       // [B,S,DK]
    float* attn = out + (size_t)nB * SEQ * DK;         // [B,S,S]

    dim3 grid(nB * (SEQ / 16));   // 8192 workgroups: one per 16-row query tile
    dim3 block(BDIM);             // 4 wave32
    sdpa_wmma_f32_kernel<<<grid, block, 0, stream>>>(q, k, v, mask, out, attn);
}